// CTCLoss_43138651521306
// MI455X (gfx1250) — compile-verified
//
#include <hip/hip_runtime.h>
#include <hip/hip_bf16.h>

// Problem constants (from reference setup_inputs): acts [T,B,V], labels [B*Lmax]
#define CTC_T    500
#define CTC_B    32
#define CTC_V    2000
#define CTC_LMAX 100
#define CTC_S    (2 * CTC_LMAX + 1)   // 201
#define NEG_INF  (-1.0e30f)

typedef __attribute__((ext_vector_type(2))) float v2f;
typedef __attribute__((ext_vector_type(8))) float v8f;

// ---------------------------------------------------------------------------
// Kernel 1: per-row logsumexp over V=2000, online (single pass) max+sum.
// Wave32 layout: each wave handles 16 consecutive rows of acts.
//   lane L (0..31): row = base + (L & 15), column half = (L >> 4) (1000 cols).
// Each lane keeps 2 partial exp-sums (s0 = even slots, s1 = odd slots) so the
// per-row 4-way reduction {s0,s1}(lane r) x {s0,s1}(lane r+16) is exactly the
// 16x4 f32 A-matrix of V_WMMA_F32_16X16X4_F32; multiplying by an all-ones
// 4x16 B matrix yields all 16 row sums in one tensor op.
// ---------------------------------------------------------------------------
__global__ void __launch_bounds__(256)
ctc_lse_kernel(const float* __restrict__ acts, float* __restrict__ lse_out) {
    const int lane = threadIdx.x & 31;
    const int wave = threadIdx.x >> 5;
    const int r16  = lane & 15;
    const int half = lane >> 4;
    const int rowBase = blockIdx.x * 128 + wave * 16;   // 8 waves * 16 rows
    const int row     = rowBase + r16;                  // grid sized exactly

    const float4* p =
        (const float4*)(acts + (size_t)row * CTC_V + (size_t)half * (CTC_V / 2));

    // Online logsumexp over this lane's 1000 contiguous elements.
    float m  = -3.0e38f;
    float s0 = 0.0f, s1 = 0.0f;
    #pragma unroll 4
    for (int i = 0; i < (CTC_V / 2) / 4; ++i) {   // 250 float4 loads
        float4 x = p[i];
        float mx = fmaxf(fmaxf(x.x, x.y), fmaxf(x.z, x.w));
        float mn = fmaxf(m, mx);
        float sc = __expf(m - mn);
        s0 = s0 * sc + __expf(x.x - mn) + __expf(x.z - mn);
        s1 = s1 * sc + __expf(x.y - mn) + __expf(x.w - mn);
        m = mn;
    }

    // Unify the running max between the two halves of each row, rescale sums.
    float mo   = __shfl_xor(m, 16, 32);
    float mrow = fmaxf(m, mo);
    float sc   = __expf(m - mrow);
    s0 *= sc;
    s1 *= sc;

    // WMMA row-sum: A[16x4] = partial sums, B[4x16] = ones, D[r][*] = rowsum(r).
    v2f a;  a.x = s0;   a.y = s1;
    v2f bo; bo.x = 1.0f; bo.y = 1.0f;
    v8f c = {};
    c = __builtin_amdgcn_wmma_f32_16x16x4_f32(
            /*neg_a=*/false, a, /*neg_b=*/false, bo,
            /*c_mod=*/(short)0, c, /*reuse_a=*/false, /*reuse_b=*/false);

    // D layout: VGPR j -> lanes 0-15: M=j ; lanes 16-31: M=j+8 (N replicated).
    __shared__ float sums[8][16];
    if (lane == 0) {
        sums[wave][0] = c[0]; sums[wave][1] = c[1];
        sums[wave][2] = c[2]; sums[wave][3] = c[3];
        sums[wave][4] = c[4]; sums[wave][5] = c[5];
        sums[wave][6] = c[6]; sums[wave][7] = c[7];
    } else if (lane == 16) {
        sums[wave][8]  = c[0]; sums[wave][9]  = c[1];
        sums[wave][10] = c[2]; sums[wave][11] = c[3];
        sums[wave][12] = c[4]; sums[wave][13] = c[5];
        sums[wave][14] = c[6]; sums[wave][15] = c[7];
    }
    __syncthreads();

    if (half == 0) {
        float srow = sums[wave][r16];        // sum for row r16; mrow matches it
        lse_out[row] = mrow + __logf(srow);
    }
}

// ---------------------------------------------------------------------------
// Kernel 2: CTC alpha recursion. One block per batch element, thread s = state.
// alpha double-buffered in LDS, one barrier per timestep; next timestep's
// gathered logit + lse prefetched into registers to hide gather latency.
// ---------------------------------------------------------------------------
__global__ void __launch_bounds__(256)
ctc_dp_kernel(const float* __restrict__ acts, const int* __restrict__ labels,
              const int* __restrict__ act_lens, const int* __restrict__ label_lens,
              const float* __restrict__ lse, float* __restrict__ out) {
    const int b = blockIdx.x;
    const int s = threadIdx.x;

    __shared__ float bufA[CTC_S];
    __shared__ float bufB[CTC_S];

    // Extended label for state s: blank(0) on even s, labels[(s-1)/2] on odd s.
    int lab = 0, labprev = -1;
    if (s < CTC_S && (s & 1)) {
        lab     = labels[b * CTC_LMAX + (s >> 1)];
        labprev = (s >= 3) ? labels[b * CTC_LMAX + (s >> 1) - 1] : -1;
    }
    const int  ext_s    = lab;
    const bool can_skip = (s & 1) && (lab != 0) && (lab != labprev);

    const int  act_len = act_lens[b];
    const int  lab_len = label_lens[b];
    const bool svalid  = s < (2 * lab_len + 1);

    // t = 0 init
    float a0 = NEG_INF;
    if (s < CTC_S) {
        float g0 = acts[(size_t)b * CTC_V + ext_s] - lse[b];
        if (s == 0)                      a0 = g0;
        else if (s == 1 && lab_len > 0)  a0 = g0;
    }
    float* cur = bufA;
    float* nxt = bufB;
    if (s < CTC_S) cur[s] = a0;
    __syncthreads();

    const int Tl = (act_len < CTC_T) ? act_len : CTC_T;  // frozen past act_len

    // Prefetch t = 1
    float gn = 0.0f, ln = 0.0f;
    if (Tl > 1) {
        if (s < CTC_S) gn = acts[((size_t)1 * CTC_B + b) * CTC_V + ext_s];
        ln = lse[1 * CTC_B + b];
    }

    for (int t = 1; t < Tl; ++t) {
        float gt = gn - ln;
        if (t + 1 < Tl) {   // software-pipelined gather for t+1
            if (s < CTC_S) gn = acts[((size_t)(t + 1) * CTC_B + b) * CTC_V + ext_s];
            ln = lse[(t + 1) * CTC_B + b];
        }
        if (s < CTC_S) {
            float a  = cur[s];
            float a1 = (s >= 1) ? cur[s - 1] : NEG_INF;
            float a2 = (s >= 2 && can_skip) ? cur[s - 2] : NEG_INF;
            float mm  = fmaxf(a, fmaxf(a1, a2));
            float sum = __expf(a - mm) + __expf(a1 - mm) + __expf(a2 - mm);
            float na  = mm + __logf(sum) + gt;
            nxt[s] = svalid ? na : NEG_INF;
        }
        __syncthreads();
        float* tmp = cur; cur = nxt; nxt = tmp;
    }

    if (s == 0) {
        int   end = 2 * lab_len;
        float ae  = cur[end];
        float ap  = (lab_len > 0) ? cur[end - 1] : NEG_INF;
        float mm  = fmaxf(ae, ap);
        float loss = -(mm + __logf(__expf(ae - mm) + __expf(ap - mm)));
        atomicAdd(out, loss);
    }
}

__global__ void ctc_zero_kernel(float* __restrict__ out) {
    if (threadIdx.x == 0 && blockIdx.x == 0) out[0] = 0.0f;
}

extern "C" void kernel_launch(void* const* d_in, const int* in_sizes, int n_in,
                              void* d_out, int out_size, void* d_ws, size_t ws_size,
                              hipStream_t stream) {
    const float* acts       = (const float*)d_in[0];
    const int*   labels     = (const int*)d_in[1];
    const int*   act_lens   = (const int*)d_in[2];
    const int*   label_lens = (const int*)d_in[3];
    float*       out        = (float*)d_out;
    float*       lse        = (float*)d_ws;   // T*B floats = 64 KB of scratch

    // 16000 rows, 128 rows per 256-thread block (8 waves x 16 rows) -> 125 blocks
    ctc_lse_kernel<<<(CTC_T * CTC_B) / 128, 256, 0, stream>>>(acts, lse);
    ctc_zero_kernel<<<1, 32, 0, stream>>>(out);
    ctc_dp_kernel<<<CTC_B, 256, 0, stream>>>(acts, labels, act_lens, label_lens,
                                             lse, out);
}